// EncoderLSTM_12661563589296
// MI455X (gfx1250) — compile-verified
//
#include <hip/hip_runtime.h>
#include <hip/hip_bf16.h>
#include <math.h>

// ---------------------------------------------------------------------------
// EncoderLSTM for MI455X (gfx1250), wave32.
//   T=1024, B=256, H=512, IN=3, ZX=ZY=64, ZE=16
// Persistent cooperative LSTM kernel:
//   grid = 64 WGs x 256 threads (8 waves). Wave (wg,w) owns M-tile
//   m = (wg>>5)*8 + w (16 batch rows) and h-column tile n = wg&31 (16 cols),
//   plus the 4 gate tiles at columns q*512 + n*16.  W_hh slice (4x16x512 f32)
//   staged once into LDS via global_load_async_to_lds_b128; cell state c
//   lives in 8 VGPRs/lane. One atomic grid barrier per step, h double-
//   buffered in d_ws.
// GEMM via V_WMMA_F32_16X16X4_F32 (full f32 precision, matches reference).
// Activations via hardware V_TANH_F32 (sigmoid(x) = 0.5*tanh(x/2)+0.5).
// ---------------------------------------------------------------------------

typedef __attribute__((ext_vector_type(2))) float v2f;
typedef __attribute__((ext_vector_type(8))) float v8f;

#define T_STEPS 1024
#define BATCH   256
#define HID     512
#define KSTR    516          // padded LDS row stride (floats): 2064B rows,
                             // 16B-aligned, conflict-free b64 reads
#define NWG     64
#define HFLOATS (BATCH * HID)  // floats per h buffer

__device__ __forceinline__ float htanh(float x) {
  float r;
  asm volatile("v_tanh_f32 %0, %1\n\tv_nop" : "=v"(r) : "v"(x));
  return r;
}
__device__ __forceinline__ float hsigmoid(float x) {
  return fmaf(htanh(0.5f * x), 0.5f, 0.5f);
}

__global__ void init_barrier_kernel(unsigned* bar) {
  if (threadIdx.x < 2) bar[threadIdx.x] = 0u;
}

__device__ __forceinline__ void grid_sync(unsigned* cnt, unsigned* gen) {
  __syncthreads();
  if (threadIdx.x == 0) {
    volatile unsigned* vgen = gen;
    unsigned g = *vgen;
    __threadfence();
    if (atomicAdd(cnt, 1u) == NWG - 1) {
      atomicExch(cnt, 0u);
      __threadfence();
      atomicAdd(gen, 1u);
    } else {
      while (*vgen == g) { /* spin */ }
    }
  }
  __syncthreads();
  __threadfence();
}

__global__ __launch_bounds__(256) void lstm_kernel(
    const float* __restrict__ x, const float* __restrict__ a,
    const float* __restrict__ y,
    const float* __restrict__ W_ih, const float* __restrict__ W_hh,
    const float* __restrict__ b_ih, const float* __restrict__ b_hh,
    float* __restrict__ hbuf,     // 2 * BATCH*HID floats (double buffer)
    unsigned* __restrict__ bar) {
  // W_hh slice for this WG: [gate q][row 0..15][k 0..511], padded rows.
  __shared__ float ldsW[4 * 16 * KSTR];   // 132096 bytes (<320KB/WGP)

  const int tid  = threadIdx.x;
  const int wg   = blockIdx.x;
  const int nIdx = wg & 31;              // h-column tile index (0..31)
  const int mGrp = wg >> 5;              // batch-row group (0..1)
  const int wave = tid >> 5;             // wave32
  const int lane = tid & 31;
  const int half = lane >> 4;
  const int lan  = lane & 15;
  const int m    = mGrp * 8 + wave;      // M tile (0..15)
  const int rowbase = m * 16 + half * 8; // acc element e -> row rowbase+e
  const int arow    = m * 16 + lan;      // A-fragment row for this lane
  const int ncol    = nIdx * 16 + lan;   // h output column for this lane

  // ---- one-time: stage W_hh slice into LDS via async DMA ------------------
  // 64 rows x 512 floats = 8192 16-byte chunks; each lane DMAs one chunk
  // per issued instruction (global_load_async_to_lds_b128, ASYNCcnt).
  {
    const unsigned ldsbase = (unsigned)(uintptr_t)(&ldsW[0]);
    for (int i = tid; i < 64 * 128; i += 256) {
      int r  = i >> 7;         // 0..63 = q*16 + rr
      int c4 = i & 127;        // float4 index within row
      int q  = r >> 4, rr = r & 15;
      int g  = q * 512 + nIdx * 16 + rr;          // global gate row
      unsigned lds_off = ldsbase + (unsigned)((r * KSTR + c4 * 4) * 4);
      unsigned long long gaddr =
          (unsigned long long)(uintptr_t)(W_hh + (size_t)g * 512 + c4 * 4);
      asm volatile("global_load_async_to_lds_b128 %0, %1, off"
                   :: "v"(lds_off), "v"(gaddr) : "memory");
    }
    asm volatile("s_wait_asynccnt 0x0" ::: "memory");
  }

  // per-lane gate-column constants (fixed for whole kernel)
  float bias[4], w0[4], w1[4], w2[4];
#pragma unroll
  for (int q = 0; q < 4; ++q) {
    int g = q * 512 + nIdx * 16 + lan;
    bias[q] = b_ih[g] + b_hh[g];
    w0[q] = W_ih[g * 3 + 0];
    w1[q] = W_ih[g * 3 + 1];
    w2[q] = W_ih[g * 3 + 2];
  }

  // zero initial h (buffer 0): each WG zeroes its 2048-float slice
  {
    const int slice = HFLOATS / NWG;     // 2048
    for (int i = tid; i < slice; i += 256) hbuf[wg * slice + i] = 0.0f;
  }

  __syncthreads();
  grid_sync(bar, bar + 1);

  v8f c = {};  // cell state for this wave's 16x16 block, registers for all T

  for (int s = 0; s < T_STEPS; ++s) {
    const int t = T_STEPS - 1 - s;       // reversed-sequence LSTM
    const float* __restrict__ hin  = hbuf + (s & 1) * HFLOATS;
    float* __restrict__       hout = hbuf + ((s + 1) & 1) * HFLOATS;

    // prefetch next step's input rows (global_prefetch_b8)
    if (t > 0) {
      __builtin_prefetch(x + (t - 1) * BATCH + rowbase, 0, 3);
      __builtin_prefetch(a + (t - 1) * BATCH + rowbase, 0, 3);
      __builtin_prefetch(y + (t - 1) * BATCH + rowbase, 0, 3);
    }

    // ---- accumulators = input projection + biases (IN=3, folded in) -------
    v8f acc0, acc1, acc2, acc3;
    {
      const float* xr = x + t * BATCH + rowbase;
      const float* ar = a + t * BATCH + rowbase;
      const float* yr = y + t * BATCH + rowbase;
#pragma unroll
      for (int e = 0; e < 8; ++e) {
        float xv = xr[e], av = ar[e], yv = yr[e];
        acc0[e] = bias[0] + xv * w0[0] + av * w1[0] + yv * w2[0];
        acc1[e] = bias[1] + xv * w0[1] + av * w1[1] + yv * w2[1];
        acc2[e] = bias[2] + xv * w0[2] + av * w1[2] + yv * w2[2];
        acc3[e] = bias[3] + xv * w0[3] + av * w1[3] + yv * w2[3];
      }
    }

    // ---- G += h @ W_hh^T : f32 WMMA, K swept 4 at a time -------------------
    // A (16x4): lane l, vgpr v -> M=l&15, K = v + 2*(l>>4)  => float2 load
    // B (4x16): lane l, vgpr v -> N=l&15, K = v + 2*(l>>4)  => float2 from LDS
    const float* __restrict__ hrow = hin + arow * HID;
    const float* __restrict__ bb   = ldsW + lan * KSTR;
#pragma unroll 4
    for (int k = 0; k < HID; k += 4) {
      v2f A  = *(const v2f*)(hrow + k + 2 * half);
      v2f B0 = *(const v2f*)(bb + 0 * 16 * KSTR + k + 2 * half);
      v2f B1 = *(const v2f*)(bb + 1 * 16 * KSTR + k + 2 * half);
      v2f B2 = *(const v2f*)(bb + 2 * 16 * KSTR + k + 2 * half);
      v2f B3 = *(const v2f*)(bb + 3 * 16 * KSTR + k + 2 * half);
      acc0 = __builtin_amdgcn_wmma_f32_16x16x4_f32(false, A, false, B0,
                                                   (short)0, acc0, false, false);
      acc1 = __builtin_amdgcn_wmma_f32_16x16x4_f32(false, A, false, B1,
                                                   (short)0, acc1, false, false);
      acc2 = __builtin_amdgcn_wmma_f32_16x16x4_f32(false, A, false, B2,
                                                   (short)0, acc2, false, false);
      acc3 = __builtin_amdgcn_wmma_f32_16x16x4_f32(false, A, false, B3,
                                                   (short)0, acc3, false, false);
    }

    // ---- cell update in accumulator layout (gates i,f,g,o = q 0..3) --------
    // sigmoid/tanh via hardware v_tanh_f32 (branch-free TRANS ops)
    v8f hnew;
#pragma unroll
    for (int e = 0; e < 8; ++e) {
      float ig = hsigmoid(acc0[e]);
      float fg = hsigmoid(acc1[e]);
      float gg = htanh(acc2[e]);
      float og = hsigmoid(acc3[e]);
      float cv = fg * c[e] + ig * gg;
      c[e] = cv;
      hnew[e] = og * htanh(cv);
    }

    // ---- publish h slice ---------------------------------------------------
#pragma unroll
    for (int e = 0; e < 8; ++e)
      hout[(rowbase + e) * HID + ncol] = hnew[e];

    grid_sync(bar, bar + 1);
  }
  // final h lands in buffer (T_STEPS & 1) == 0
}

// ---------------------------------------------------------------------------
// Heads: zx = h@W_xi^T + b ; ze = softmax(h@W_eta^T + b) ;
//        zy = [h, zx]@W_zeta^T + b.   One block per batch row, 64 threads.
// Output layout: zx (256*64) | zy (256*64) | ze (256*16)
// ---------------------------------------------------------------------------
__global__ __launch_bounds__(64) void heads_kernel(
    const float* __restrict__ h,
    const float* __restrict__ W_eta, const float* __restrict__ b_eta,
    const float* __restrict__ W_xi,  const float* __restrict__ b_xi,
    const float* __restrict__ W_zeta,const float* __restrict__ b_zeta,
    float* __restrict__ out) {
  const int row = blockIdx.x;
  const int tid = threadIdx.x;
  __shared__ float sh[HID + 64];
  __shared__ float le[16];

  for (int i = tid; i < HID; i += 64) sh[i] = h[row * HID + i];
  __syncthreads();

  // zx (64 cols)
  float zx = b_xi[tid];
  for (int k = 0; k < HID; ++k) zx += sh[k] * W_xi[tid * HID + k];
  out[row * 64 + tid] = zx;
  sh[HID + tid] = zx;

  // ze logits (16 cols)
  if (tid < 16) {
    float e = b_eta[tid];
    for (int k = 0; k < HID; ++k) e += sh[k] * W_eta[tid * HID + k];
    le[tid] = e;
  }
  __syncthreads();

  // zy (64 cols over concat [h, zx] of length 576)
  float zy = b_zeta[tid];
  for (int k = 0; k < HID + 64; ++k) zy += sh[k] * W_zeta[tid * (HID + 64) + k];
  out[BATCH * 64 + row * 64 + tid] = zy;

  // softmax over 16 ze logits
  if (tid < 16) {
    float mx = le[0];
#pragma unroll
    for (int i = 1; i < 16; ++i) mx = fmaxf(mx, le[i]);
    float sum = 0.0f;
#pragma unroll
    for (int i = 0; i < 16; ++i) sum += __expf(le[i] - mx);
    out[2 * BATCH * 64 + row * 16 + tid] = __expf(le[tid] - mx) / sum;
  }
}

extern "C" void kernel_launch(void* const* d_in, const int* in_sizes, int n_in,
                              void* d_out, int out_size, void* d_ws, size_t ws_size,
                              hipStream_t stream) {
  const float* x      = (const float*)d_in[0];
  const float* a      = (const float*)d_in[1];
  const float* y      = (const float*)d_in[2];
  const float* W_ih   = (const float*)d_in[3];
  const float* W_hh   = (const float*)d_in[4];
  const float* b_ih   = (const float*)d_in[5];
  const float* b_hh   = (const float*)d_in[6];
  const float* W_eta  = (const float*)d_in[7];
  const float* b_eta  = (const float*)d_in[8];
  const float* W_xi   = (const float*)d_in[9];
  const float* b_xi   = (const float*)d_in[10];
  const float* W_zeta = (const float*)d_in[11];
  const float* b_zeta = (const float*)d_in[12];
  float* out = (float*)d_out;

  // workspace layout: [0..255] barrier (count, gen), then h double buffer
  unsigned* bar = (unsigned*)d_ws;
  float* hbuf = (float*)((char*)d_ws + 256);   // 2 * 256*512 floats = 1 MB

  init_barrier_kernel<<<1, 64, 0, stream>>>(bar);
  lstm_kernel<<<NWG, 256, 0, stream>>>(x, a, y, W_ih, W_hh, b_ih, b_hh,
                                       hbuf, bar);
  heads_kernel<<<BATCH, 64, 0, stream>>>(hbuf /* final h in buffer 0 */,
                                         W_eta, b_eta, W_xi, b_xi,
                                         W_zeta, b_zeta, out);
}